// GAT_71107478553229
// MI455X (gfx1250) — compile-verified
//
#include <hip/hip_runtime.h>
#include <hip/hip_bf16.h>
#include <math.h>

typedef float v2f __attribute__((ext_vector_type(2)));
typedef float v8f __attribute__((ext_vector_type(8)));

#define N_NODES 50000
#define E_RAW   800000
#define E_TOT   (E_RAW + N_NODES)

// ---------------------------------------------------------------------------
// utility kernels
// ---------------------------------------------------------------------------
__global__ void k_fill(float* __restrict__ p, float v, int n) {
    int i = blockIdx.x * blockDim.x + threadIdx.x;
    if (i < n) p[i] = v;
}

// zero-pad W [Fin, Fout] -> Wp [Fin, WLD]
__global__ void k_padW(const float* __restrict__ W, float* __restrict__ Wp,
                       int Fin, int Fout, int WLD) {
    int i = blockIdx.x * blockDim.x + threadIdx.x;
    if (i >= Fin * WLD) return;
    int r = i / WLD, c = i - r * WLD;
    Wp[i] = (c < Fout) ? W[r * Fout + c] : 0.0f;
}

// ---------------------------------------------------------------------------
// GEMM: Z[nRows, WLD] = X[nRows, Fin] @ W[Fin, WLD], fp32 WMMA 16x16x4.
// Requirements (guaranteed by the launcher): nRows % 16 == 0, WLD % 64 == 0,
// W zero-padded to WLD columns. No guards anywhere -> clean inner loop.
// One wave computes a 16x64 tile; 4 accumulators reuse each A fragment.
// Lane layout per CDNA5 ISA 7.12.2:
//   A (16x4 f32):  lane L -> M = L&15 ; vgpr j holds K = 2*(L>>4)+j
//   B (4x16 f32):  lane L -> N = L&15 ; vgpr j holds K = 2*(L>>4)+j
//   C/D (16x16):   vgpr r -> M = r + 8*(L>>4), N = L&15
// ---------------------------------------------------------------------------
__global__ void k_gemm_wmma(const float* __restrict__ X,
                            const float* __restrict__ W,
                            float* __restrict__ Z,
                            int Fin, int WLD, int ctiles) {
    int tile = blockIdx.x;
    int ct   = tile % ctiles;
    int rt   = tile / ctiles;
    int lane = threadIdx.x & 31;
    int half = lane >> 4;          // 0 or 1
    int l15  = lane & 15;
    int kk   = 2 * half;           // K offset within 4-step for this half-wave

    const float* xrow = X + (size_t)(rt * 16 + l15) * Fin + kk;
    const float* wptr = W + (size_t)kk * WLD + ct * 64 + l15;
    size_t wstep = (size_t)4 * WLD;

    v8f c0 = {}, c1 = {}, c2 = {}, c3 = {};

#pragma unroll 4
    for (int k = 0; k < Fin; k += 4) {
        v2f a  = { xrow[0], xrow[1] };
        v2f b0 = { wptr[0],  wptr[WLD + 0]  };
        v2f b1 = { wptr[16], wptr[WLD + 16] };
        v2f b2 = { wptr[32], wptr[WLD + 32] };
        v2f b3 = { wptr[48], wptr[WLD + 48] };

        c0 = __builtin_amdgcn_wmma_f32_16x16x4_f32(false, a, false, b0, (short)0, c0, false, false);
        c1 = __builtin_amdgcn_wmma_f32_16x16x4_f32(false, a, false, b1, (short)0, c1, false, false);
        c2 = __builtin_amdgcn_wmma_f32_16x16x4_f32(false, a, false, b2, (short)0, c2, false, false);
        c3 = __builtin_amdgcn_wmma_f32_16x16x4_f32(false, a, false, b3, (short)0, c3, false, false);

        xrow += 4;
        wptr += wstep;
    }

    // store: VGPR r holds row rt*16 + r + 8*half, column ct*64 + {0,16,32,48} + l15
    float* zp = Z + (size_t)(rt * 16 + 8 * half) * WLD + ct * 64 + l15;
#pragma unroll
    for (int r = 0; r < 8; ++r) {
        zp[0]  = c0[r];
        zp[16] = c1[r];
        zp[32] = c2[r];
        zp[48] = c3[r];
        zp += WLD;
    }
}

// ---------------------------------------------------------------------------
// per-node attention projections: el[n,h] = sum_f z[n,h,f]*al[h,f]; er likewise
// z has leading dimension zld (valid cols 0..H*F-1 contiguous)
// ---------------------------------------------------------------------------
__global__ void k_attn(const float* __restrict__ z,
                       const float* __restrict__ al,
                       const float* __restrict__ ar,
                       float* __restrict__ el, float* __restrict__ er,
                       int N, int H, int F, int zld) {
    int i = blockIdx.x * blockDim.x + threadIdx.x;
    if (i >= N * H) return;
    int n = i / H, h = i - n * H;
    const float* zp  = z  + (size_t)n * zld + (size_t)h * F;
    const float* alp = al + (size_t)h * F;
    const float* arp = ar + (size_t)h * F;
    float sl = 0.0f, sr = 0.0f;
    for (int f = 0; f < F; ++f) {
        float v = zp[f];
        sl += v * alp[f];
        sr += v * arp[f];
    }
    el[i] = sl;
    er[i] = sr;
}

// ---------------------------------------------------------------------------
// edge helpers (implicit self-loops appended at indices >= E_RAW)
// ---------------------------------------------------------------------------
__device__ __forceinline__ void edge_sd(int e, const int* __restrict__ src,
                                        const int* __restrict__ dst,
                                        int& s, int& d, bool& inv) {
    if (e < E_RAW) { s = src[e]; d = dst[e]; inv = (s == d); }
    else           { s = d = e - E_RAW;      inv = false;    }
}

__device__ __forceinline__ float atomicMaxF32(float* addr, float v) {
    if (v >= 0.0f)
        return __int_as_float(atomicMax((int*)addr, __float_as_int(v)));
    else
        return __uint_as_float(atomicMin((unsigned int*)addr, __float_as_uint(v)));
}

// pass 1: e = leakyrelu(el[src]+er[dst]); segment max into mbuf[dst,h]
__global__ void k_edge_e(const int* __restrict__ src, const int* __restrict__ dst,
                         const float* __restrict__ el, const float* __restrict__ er,
                         float* __restrict__ ebuf, float* __restrict__ mbuf, int H) {
    int i = blockIdx.x * blockDim.x + threadIdx.x;
    if (i >= E_TOT * H) return;
    int e = i / H, h = i - e * H;
    int s, d; bool inv;
    edge_sd(e, src, dst, s, d, inv);
    float v = el[s * H + h] + er[d * H + h];
    v = (v > 0.0f) ? v : 0.2f * v;           // LeakyReLU(0.2)
    if (inv) {
        ebuf[i] = -INFINITY;
    } else {
        ebuf[i] = v;
        atomicMaxF32(&mbuf[d * H + h], v);
    }
}

// pass 2: ex = exp(e - m[dst]); segment sum into sbuf[dst,h]; ebuf <- ex
__global__ void k_edge_exp(const int* __restrict__ src, const int* __restrict__ dst,
                           float* __restrict__ ebuf,
                           const float* __restrict__ mbuf, float* __restrict__ sbuf,
                           int H) {
    int i = blockIdx.x * blockDim.x + threadIdx.x;
    if (i >= E_TOT * H) return;
    int e = i / H, h = i - e * H;
    int s, d; bool inv;
    edge_sd(e, src, dst, s, d, inv);
    float ex = 0.0f;
    if (!inv) ex = expf(ebuf[i] - mbuf[d * H + h]);
    ebuf[i] = ex;
    if (ex != 0.0f) atomicAdd(&sbuf[d * H + h], ex);
}

// pass 3: acc[dst,h,f] += z[src,h,f] * (ex / sum[dst,h]); one thread per (e,h,f)
__global__ void k_edge_msg(const int* __restrict__ src, const int* __restrict__ dst,
                           const float* __restrict__ ebuf, const float* __restrict__ sbuf,
                           const float* __restrict__ z, float* __restrict__ acc,
                           int H, int F, int zld) {
    int i = blockIdx.x * blockDim.x + threadIdx.x;
    int HF = H * F;
    if (i >= E_TOT * HF) return;
    int e = i / HF, r = i - e * HF;
    int h = r / F;
    float ex = ebuf[e * H + h];
    if (ex == 0.0f) return;
    int s, d; bool inv;
    edge_sd(e, src, dst, s, d, inv);
    float alpha = ex / sbuf[d * H + h];
    atomicAdd(&acc[(size_t)d * HF + r], z[(size_t)s * zld + r] * alpha);
}

// bias + ELU, written to dstp (may alias acc)
__global__ void k_finish(const float* __restrict__ acc, const float* __restrict__ bias,
                         float* __restrict__ dstp, int N, int HF) {
    int i = blockIdx.x * blockDim.x + threadIdx.x;
    if (i >= N * HF) return;
    float v = acc[i] + bias[i % HF];
    dstp[i] = (v > 0.0f) ? v : expm1f(v);
}

// ---------------------------------------------------------------------------
// launch
// ---------------------------------------------------------------------------
static inline int cdiv(long long a, long long b) { return (int)((a + b - 1) / b); }

extern "C" void kernel_launch(void* const* d_in, const int* in_sizes, int n_in,
                              void* d_out, int out_size, void* d_ws, size_t ws_size,
                              hipStream_t stream) {
    (void)in_sizes; (void)n_in; (void)out_size; (void)ws_size;

    const float* x   = (const float*)d_in[0];
    const int*   src = (const int*)  d_in[1];
    const int*   dst = (const int*)  d_in[2];
    const float* W0  = (const float*)d_in[3];
    const float* al0 = (const float*)d_in[4];
    const float* ar0 = (const float*)d_in[5];
    const float* b0  = (const float*)d_in[6];
    const float* W1  = (const float*)d_in[7];
    const float* al1 = (const float*)d_in[8];
    const float* ar1 = (const float*)d_in[9];
    const float* b1  = (const float*)d_in[10];
    float* out = (float*)d_out;

    const int N = N_NODES;
    const int FIN0 = 256, H0 = 4, F0 = 32, FOUT0 = H0 * F0;   // 128, WLD0 = 128
    const int FIN1 = 128, H1 = 1, F1 = 40, FOUT1 = H1 * F1;   // 40,  WLD1 = 64
    const int WLD0 = 128, WLD1 = 64;

    // workspace layout (floats)
    float* ws   = (float*)d_ws;
    float* z    = ws;                         // N*128 (layer0 z; layer1 z uses N*64)
    float* acc  = z    + (size_t)N * 128;     // N*128 (layer0 accum -> h0 in place)
    float* el   = acc  + (size_t)N * 128;     // N*4
    float* er   = el   + (size_t)N * 4;       // N*4
    float* mbuf = er   + (size_t)N * 4;       // N*4
    float* sbuf = mbuf + (size_t)N * 4;       // N*4
    float* ebuf = sbuf + (size_t)N * 4;       // E_TOT*4
    float* W1p  = ebuf + (size_t)E_TOT * 4;   // 128*64 zero-padded W1

    const int B = 256;
    const int rtiles = N / 16;                // 3125 (exact)

    // ================= layer 0 =================
    k_fill<<<cdiv((long long)N * H0, B), B, 0, stream>>>(mbuf, -INFINITY, N * H0);
    k_fill<<<cdiv((long long)N * H0, B), B, 0, stream>>>(sbuf, 0.0f, N * H0);
    k_fill<<<cdiv((long long)N * FOUT0, B), B, 0, stream>>>(acc, 0.0f, N * FOUT0);

    // z = x @ W0  (WMMA f32, Fout = 128 = 2x64, no padding needed)
    k_gemm_wmma<<<rtiles * (WLD0 / 64), 32, 0, stream>>>(x, W0, z, FIN0, WLD0, WLD0 / 64);

    k_attn<<<cdiv((long long)N * H0, B), B, 0, stream>>>(z, al0, ar0, el, er, N, H0, F0, WLD0);
    k_edge_e  <<<cdiv((long long)E_TOT * H0, B), B, 0, stream>>>(src, dst, el, er, ebuf, mbuf, H0);
    k_edge_exp<<<cdiv((long long)E_TOT * H0, B), B, 0, stream>>>(src, dst, ebuf, mbuf, sbuf, H0);
    k_edge_msg<<<cdiv((long long)E_TOT * H0 * F0, B), B, 0, stream>>>(src, dst, ebuf, sbuf, z, acc, H0, F0, WLD0);
    k_finish<<<cdiv((long long)N * FOUT0, B), B, 0, stream>>>(acc, b0, acc, N, FOUT0);  // h0 in place

    // ================= layer 1 =================
    k_fill<<<cdiv((long long)N * H1, B), B, 0, stream>>>(mbuf, -INFINITY, N * H1);
    k_fill<<<cdiv((long long)N * H1, B), B, 0, stream>>>(sbuf, 0.0f, N * H1);
    k_fill<<<cdiv((long long)N * FOUT1, B), B, 0, stream>>>(out, 0.0f, N * FOUT1);

    // pad W1 [128,40] -> W1p [128,64]
    k_padW<<<cdiv((long long)FIN1 * WLD1, B), B, 0, stream>>>(W1, W1p, FIN1, FOUT1, WLD1);

    // z1 = h0 @ W1p  (WMMA f32, padded Z leading dim 64)
    k_gemm_wmma<<<rtiles * (WLD1 / 64), 32, 0, stream>>>(acc, W1p, z, FIN1, WLD1, WLD1 / 64);

    k_attn<<<cdiv((long long)N * H1, B), B, 0, stream>>>(z, al1, ar1, el, er, N, H1, F1, WLD1);
    k_edge_e  <<<cdiv((long long)E_TOT * H1, B), B, 0, stream>>>(src, dst, el, er, ebuf, mbuf, H1);
    k_edge_exp<<<cdiv((long long)E_TOT * H1, B), B, 0, stream>>>(src, dst, ebuf, mbuf, sbuf, H1);
    k_edge_msg<<<cdiv((long long)E_TOT * H1 * F1, B), B, 0, stream>>>(src, dst, ebuf, sbuf, z, out, H1, F1, WLD1);
    // bias + ELU into d_out; mean over H1=1 heads is identity
    k_finish<<<cdiv((long long)N * FOUT1, B), B, 0, stream>>>(out, b1, out, N, FOUT1);
}